// MultiplexNodeFeatureEncoder_22428319220158
// MI455X (gfx1250) — compile-verified
//
#include <hip/hip_runtime.h>

typedef float v2f __attribute__((ext_vector_type(2)));
typedef float v8f __attribute__((ext_vector_type(8)));

#define MT        64          // rows per workgroup
#define WAVES     4
#define THREADS   128
#define FTOT      132         // input feature count (4 type cols + 128)
#define HDIM      512         // T*D
#define DOUT      128
#define FP        132         // feats LDS pitch (floats) -> bank stride 4
#define HP        516         // h LDS pitch (floats)     -> bank stride 4
#define HBYTES    (WAVES * 16 * HP * 4)   // 132096 B
#define TYPOFF    HBYTES
#define SMEM_BYTES (HBYTES + MT * 4)      // 132352 B  (2 blocks / 320KB WGP)

__device__ __forceinline__ float gelu_exact(float x) {
    // gelu(approximate=False) = 0.5 x (1 + erf(x/sqrt(2)))
    return 0.5f * x * (1.0f + erff(x * 0.70710678118654752f));
}

__global__ __launch_bounds__(THREADS)
void mpx_encoder_kernel(const float* __restrict__ nf,   // (M,132)
                        const float* __restrict__ W1,   // (128,512)
                        const float* __restrict__ b1,   // (512)
                        const float* __restrict__ W2,   // (512,512)
                        const float* __restrict__ b2,   // (512)
                        float* __restrict__ out,        // (M,128)
                        int Mtot)
{
    extern __shared__ char smem[];
    float* feats = (float*)smem;            // [64][132] — dead after A-frag preload
    float* hbuf  = (float*)smem;            // [4][16][516] — reuses feats region
    int*   typ   = (int*)(smem + TYPOFF);   // [64]

    const int  tid  = threadIdx.x;
    const int  lane = tid & 31;
    const int  wv   = tid >> 5;     // wave id 0..3
    const int  lm   = lane & 15;    // M (A) / N (B,C,D) index
    const int  lh   = lane >> 4;    // half-wave selector
    const long base = (long)blockIdx.x * MT;

    // ---- Phase 0: stage 64x128 feature tile (coalesced b128), node types ----
    #pragma unroll
    for (int it = 0; it < 16; ++it) {
        int i  = tid + THREADS * it;        // 0..2047 float4 slots
        int r  = i >> 5;                    // row in tile
        int c4 = i & 31;                    // float4 column
        long rs = base + r; if (rs >= Mtot) rs = Mtot - 1;   // clamp (exact fit for 200000)
        const float4 v = *(const float4*)(nf + rs * FTOT + 4 + c4 * 4);
        float* d = feats + r * FP + c4 * 4;
        d[0] = v.x; d[1] = v.y; d[2] = v.z; d[3] = v.w;
    }
    if (tid < MT) {
        long rs = base + tid; if (rs >= Mtot) rs = Mtot - 1;
        const float* p = nf + rs * FTOT;
        float best = p[0]; int bi = 0;
        #pragma unroll
        for (int i = 1; i < 4; ++i) { float q = p[i]; if (q > best) { best = q; bi = i; } }
        typ[tid] = bi;
    }
    __syncthreads();

    // ---- Preload GEMM1 A fragments: whole 16x128 strip in 64 VGPRs ----
    // A(16x4 f32) layout: lane = 16*(K>=2)+M, vgpr v = K%2  -> float2 per lane
    const float* myfeat = feats + wv * 16 * FP;
    v2f afr[32];
    #pragma unroll
    for (int ks = 0; ks < 32; ++ks)
        afr[ks] = *(const v2f*)(myfeat + lm * FP + ks * 4 + lh * 2);
    __syncthreads();   // feats region now dead -> becomes hbuf

    // ---- GEMM1: h = gelu(X @ W1 + b1), one 16x512 strip per wave ----
    float* myh = hbuf + wv * 16 * HP;
    for (int j = 0; j < 32; ++j) {
        float bias = b1[j * 16 + lm];       // bias varies along N only
        v8f c;
        #pragma unroll
        for (int v = 0; v < 8; ++v) c[v] = bias;
        #pragma unroll
        for (int ks = 0; ks < 32; ++ks) {
            // B(4x16) layout: vgpr v holds row (v + 2*lh), col = lm
            const float* wp = W1 + (ks * 4 + lh * 2) * HDIM + j * 16 + lm;
            v2f b; b.x = wp[0]; b.y = wp[HDIM];
            c = __builtin_amdgcn_wmma_f32_16x16x4_f32(false, afr[ks], false, b,
                                                      (short)0, c, false, false);
        }
        #pragma unroll
        for (int v = 0; v < 8; ++v) {       // D: row = v + 8*lh, col = j*16+lm
            int row = v + lh * 8;
            myh[row * HP + j * 16 + lm] = gelu_exact(c[v]);
        }
    }
    __syncthreads();

    // ---- GEMM2 + fused type-select store: out = (h @ W2 + b2)[:, t*128 : t*128+128] ----
    for (int j = 0; j < 32; ++j) {
        float bias = b2[j * 16 + lm];
        v8f c;
        #pragma unroll
        for (int v = 0; v < 8; ++v) c[v] = bias;
        #pragma unroll 4
        for (int ks = 0; ks < 128; ++ks) {
            v2f a = *(const v2f*)(myh + lm * HP + ks * 4 + lh * 2);
            const float* wp = W2 + (ks * 4 + lh * 2) * HDIM + j * 16 + lm;
            v2f b; b.x = wp[0]; b.y = wp[HDIM];
            c = __builtin_amdgcn_wmma_f32_16x16x4_f32(false, a, false, b,
                                                      (short)0, c, false, false);
        }
        int col = j * 16 + lm;
        #pragma unroll
        for (int v = 0; v < 8; ++v) {
            int row = v + lh * 8;
            int t   = typ[wv * 16 + row];
            int c0  = t << 7;                       // t * 128
            long rg = base + wv * 16 + row;
            if ((unsigned)(col - c0) < 128u && rg < Mtot)
                out[rg * DOUT + (col - c0)] = c[v];
        }
    }
}

extern "C" void kernel_launch(void* const* d_in, const int* in_sizes, int n_in,
                              void* d_out, int out_size, void* d_ws, size_t ws_size,
                              hipStream_t stream) {
    const float* nf = (const float*)d_in[0];
    const float* W1 = (const float*)d_in[1];
    const float* b1 = (const float*)d_in[2];
    const float* W2 = (const float*)d_in[3];
    const float* b2 = (const float*)d_in[4];
    float* out = (float*)d_out;

    int Mtot = in_sizes[0] / FTOT;              // 200000
    int grid = (Mtot + MT - 1) / MT;            // 3125

    // Dynamic LDS (132 KB) exceeds the default cap; raise it (capture-safe, not a stream op).
    hipFuncSetAttribute((const void*)mpx_encoder_kernel,
                        hipFuncAttributeMaxDynamicSharedMemorySize, SMEM_BYTES);

    mpx_encoder_kernel<<<grid, THREADS, SMEM_BYTES, stream>>>(nf, W1, b1, W2, b2, out, Mtot);
}